// LightGCN_63720134803714
// MI455X (gfx1250) — compile-verified
//
#include <hip/hip_runtime.h>

typedef __attribute__((ext_vector_type(2))) float v2f;
typedef __attribute__((ext_vector_type(8))) float v8f;

#define D 32

// ---------------------------------------------------------------------------
// Kernel 1: zero the accumulator workspace (sums + degrees), b128 stores.
// ---------------------------------------------------------------------------
__global__ void zero_ws_kernel(float* __restrict__ p, size_t n) {
    size_t i = ((size_t)blockIdx.x * blockDim.x + threadIdx.x) * 4;
    size_t stride = (size_t)gridDim.x * blockDim.x * 4;
    for (; i + 3 < n; i += stride) {
        *(float4*)(p + i) = float4{0.f, 0.f, 0.f, 0.f};
    }
    if (blockIdx.x == 0 && threadIdx.x == 0) {
        for (size_t t = n & ~(size_t)3; t < n; ++t) p[t] = 0.f;
    }
}

// ---------------------------------------------------------------------------
// Kernel 2: edge scatter. 32 lanes handle one edge (lane d = feature dim d).
// lane d: item_sum[dst][d] += user_embed[src][d]; user_sum[src][d] += item_embed[dst][d]
// lane 0: degree counters. Native global_atomic_add_f32 rides the 192MB L2.
// ---------------------------------------------------------------------------
__global__ void scatter_edges_kernel(const float* __restrict__ user_embed,
                                     const float* __restrict__ item_embed,
                                     const int* __restrict__ edge_src,
                                     const int* __restrict__ edge_dst,
                                     float* __restrict__ user_sum,
                                     float* __restrict__ item_sum,
                                     float* __restrict__ user_deg,
                                     float* __restrict__ item_deg,
                                     int E) {
    int gid = blockIdx.x * blockDim.x + threadIdx.x;
    int e = gid >> 5;
    int d = gid & 31;
    if (e >= E) return;
    int s = edge_src[e];
    int t = edge_dst[e];
    float uv = user_embed[(size_t)s * D + d];
    float iv = item_embed[(size_t)t * D + d];
    unsafeAtomicAdd(&item_sum[(size_t)t * D + d], uv);
    unsafeAtomicAdd(&user_sum[(size_t)s * D + d], iv);
    if (d == 0) {
        unsafeAtomicAdd(&item_deg[t], 1.0f);
        unsafeAtomicAdd(&user_deg[s], 1.0f);
    }
}

// Fast reciprocal of a small positive integer-valued float:
// hardware v_rcp_f32 + one Newton-Raphson step (exact to <1ulp here).
__device__ __forceinline__ float fast_rcp(float x) {
    float r = __builtin_amdgcn_rcpf(x);
    r = r * (2.0f - x * r);
    return r;
}

// ---------------------------------------------------------------------------
// Kernel 3: per-node output via V_WMMA_F32_16X16X4_F32.
// One wave32 computes a 16x32 output tile:
//   out = self(16x32) @ W_self(32x32) + (sum/deg)(16x32) @ W_neigh(32x32) + b
// as two 16x16 C tiles, 16 WMMA K-steps each (8 self + 8 neigh).
//
// f32 WMMA operand layout (ISA 7.12.2):
//   A 16x4: lanes 0-15 = rows M, VGPR0/1 = K0/K1; lanes 16-31 = K2/K3.
//   B 4x16: lanes 0-15 = cols N, VGPR0/1 = K0/K1; lanes 16-31 = K2/K3.
//   C 16x16: VGPR r -> M=r (lanes 0-15) / M=r+8 (lanes 16-31), N = lane&15.
// Weights are staged transposed in LDS so B fragments are contiguous b64 loads.
// NU and NI are both multiples of 16 here (1M each), so waves never straddle
// the user/item boundary and EXEC stays all-ones for every WMMA.
// ---------------------------------------------------------------------------
__global__ __launch_bounds__(256) void sage_out_kernel(
        const float* __restrict__ user_embed,
        const float* __restrict__ item_embed,
        const float* __restrict__ W_self,    // [32,32] row-major W[k][n]
        const float* __restrict__ W_neigh,   // [32,32]
        const float* __restrict__ bias,      // [32]
        const float* __restrict__ user_sum,
        const float* __restrict__ item_sum,
        const float* __restrict__ user_deg,
        const float* __restrict__ item_deg,
        float* __restrict__ out,             // [(NU+NI), 32], user rows first
        int NU, int NT) {
    // LDS: both weight matrices transposed -> wT[mat][n*32 + k] = W[k*32 + n]
    __shared__ float wT[2][D * D];
    for (int i = threadIdx.x; i < D * D; i += blockDim.x) {
        int k = i >> 5, n = i & 31;
        wT[0][n * D + k] = W_self[i];
        wT[1][n * D + k] = W_neigh[i];
    }
    __syncthreads();

    int wave = blockIdx.x * (blockDim.x >> 5) + (threadIdx.x >> 5);
    int lane = threadIdx.x & 31;
    long base = (long)wave * 16;
    if (base >= (long)NT) return;           // wave-uniform exit

    bool isUser = base < (long)NU;
    const float* self_emb = isUser ? user_embed : item_embed;
    const float* nsum     = isUser ? user_sum  : item_sum;
    const float* ndeg     = isUser ? user_deg  : item_deg;
    long lbase = isUser ? base : base - NU;

    int m = lane & 15;        // row within tile (A) / col within tile (B,C)
    int h = lane >> 4;        // K half-select within a 4-wide K step
    long row = lbase + m;
    const float* srow = self_emb + row * D;
    const float* nrow = nsum + row * D;
    float inv = fast_rcp(fmaxf(ndeg[row], 1.0f));

    // A fragments: a[s] holds K = 4s + 2h + {0,1} of this lane's row.
    v2f a_s[8], a_n[8];
#pragma unroll
    for (int s = 0; s < 8; ++s) {
        int c0 = 4 * s + 2 * h;
        a_s[s] = *(const v2f*)(srow + c0);
        v2f an = *(const v2f*)(nrow + c0);
        an.x *= inv;
        an.y *= inv;
        a_n[s] = an;
    }

    float* otile = out + base * D;
#pragma unroll
    for (int t = 0; t < 2; ++t) {          // two 16-wide N tiles
        int col = 16 * t + m;
        float bb = bias[col];
        v8f c = {bb, bb, bb, bb, bb, bb, bb, bb};
        const float* wSelfCol  = &wT[0][col * D];
        const float* wNeighCol = &wT[1][col * D];
#pragma unroll
        for (int s = 0; s < 8; ++s) {
            v2f bfrag = *(const v2f*)(wSelfCol + 4 * s + 2 * h);
            c = __builtin_amdgcn_wmma_f32_16x16x4_f32(
                    false, a_s[s], false, bfrag, (short)0, c, false, false);
        }
#pragma unroll
        for (int s = 0; s < 8; ++s) {
            v2f bfrag = *(const v2f*)(wNeighCol + 4 * s + 2 * h);
            c = __builtin_amdgcn_wmma_f32_16x16x4_f32(
                    false, a_n[s], false, bfrag, (short)0, c, false, false);
        }
#pragma unroll
        for (int r = 0; r < 8; ++r) {
            int mrow = (h == 0) ? r : (r + 8);
            otile[(long)mrow * D + col] = c[r];
        }
    }
}

// ---------------------------------------------------------------------------
// Host launcher.
// Inputs: 0 user_embed, 1 item_embed, 2 W_self, 3 W_neigh, 4 b,
//         5 edge_src, 6 edge_dst, 7 num_user, 8 num_item
// Workspace (floats): user_sum[NU*32] | item_sum[NI*32] | user_deg[NU] | item_deg[NI]
// ---------------------------------------------------------------------------
extern "C" void kernel_launch(void* const* d_in, const int* in_sizes, int n_in,
                              void* d_out, int out_size, void* d_ws, size_t ws_size,
                              hipStream_t stream) {
    const float* user_embed = (const float*)d_in[0];
    const float* item_embed = (const float*)d_in[1];
    const float* W_self     = (const float*)d_in[2];
    const float* W_neigh    = (const float*)d_in[3];
    const float* bias       = (const float*)d_in[4];
    const int*   edge_src   = (const int*)d_in[5];
    const int*   edge_dst   = (const int*)d_in[6];

    int NU = in_sizes[0] / D;
    int NI = in_sizes[1] / D;
    int E  = in_sizes[5];

    float* ws       = (float*)d_ws;
    float* user_sum = ws;
    float* item_sum = user_sum + (size_t)NU * D;
    float* user_deg = item_sum + (size_t)NI * D;
    float* item_deg = user_deg + (size_t)NU;
    size_t ws_floats = (size_t)(D + 1) * ((size_t)NU + NI);

    // 1) clear accumulators
    zero_ws_kernel<<<2048, 256, 0, stream>>>(ws, ws_floats);

    // 2) scatter-add over edges (32 lanes per edge)
    long threadsE = (long)E * 32;
    int blocksE = (int)((threadsE + 255) / 256);
    scatter_edges_kernel<<<blocksE, 256, 0, stream>>>(
        user_embed, item_embed, edge_src, edge_dst,
        user_sum, item_sum, user_deg, item_deg, E);

    // 3) finalize: mean + two GEMMs + bias via f32 WMMA (16 nodes / wave)
    long NT = (long)NU + NI;
    long waves = (NT + 15) / 16;
    int blocksF = (int)((waves + 7) / 8);   // 8 waves (256 threads) per block
    sage_out_kernel<<<blocksF, 256, 0, stream>>>(
        user_embed, item_embed, W_self, W_neigh, bias,
        user_sum, item_sum, user_deg, item_deg,
        (float*)d_out, NU, (int)NT);
}